// MambaLayer_8065948582496
// MI455X (gfx1250) — compile-verified
//
#include <hip/hip_runtime.h>
#include <hip/hip_bf16.h>
#include <math.h>

// ---------------------------------------------------------------------------
// Bidirectional Mamba layer for MI455X (gfx1250, wave32).
// GEMMs: V_WMMA_F32_16X16X4_F32 (f32-exact; FLOP count is trivial, scan is
// the latency bottleneck). Selective scan: 16-way chunked parallel scan,
// 1 wave = 2 channels, lanes = states, shfl_xor reduction for y_t.
// ---------------------------------------------------------------------------

typedef __attribute__((ext_vector_type(2))) float v2f;
typedef __attribute__((ext_vector_type(8))) float v8f;

#define LTOK   8192          // NF*H*W
#define BATCH  4
#define BL     32768         // BATCH * LTOK
#define DM     128           // d_model
#define DI     256           // d_inner
#define NSTATE 16
#define NPROJ  40            // dt_rank + 2*d_state
#define NCH    16            // scan chunks per sequence
#define CHLEN  512           // LTOK / NCH

// ------------------------- LayerNorm 1 (x -> xn) ---------------------------
__global__ __launch_bounds__(256) void ln_in_kernel(
    const float* __restrict__ x, const float* __restrict__ w,
    const float* __restrict__ bias, float* __restrict__ xn) {
  int m = blockIdx.x * blockDim.x + threadIdx.x;   // token id
  int b = m >> 13, l = m & (LTOK - 1);
  const float* xb = x + (size_t)b * DM * LTOK + l;
  float s = 0.f, ss = 0.f;
  for (int d = 0; d < DM; ++d) { float v = xb[(size_t)d * LTOK]; s += v; ss += v * v; }
  float mean = s * (1.f / DM);
  float var  = ss * (1.f / DM) - mean * mean;
  float rs   = rsqrtf(var + 1e-5f);
  float* o = xn + (size_t)m * DM;
  for (int d = 0; d < DM; ++d) {
    float v = xb[(size_t)d * LTOK];
    o[d] = (v - mean) * rs * w[d] + bias[d];
  }
}

// ------------------ residual + LayerNorm 2 (-> d_out) ----------------------
__global__ __launch_bounds__(256) void ln_out_kernel(
    const float* __restrict__ x, const float* __restrict__ lin,
    const float* __restrict__ w, const float* __restrict__ bias,
    float* __restrict__ out) {
  int m = blockIdx.x * blockDim.x + threadIdx.x;
  int b = m >> 13, l = m & (LTOK - 1);
  const float* xb = x + (size_t)b * DM * LTOK + l;
  const float* lb = lin + (size_t)m * DM;
  float s = 0.f, ss = 0.f;
  for (int d = 0; d < DM; ++d) {
    float v = xb[(size_t)d * LTOK] + lb[d];
    s += v; ss += v * v;
  }
  float mean = s * (1.f / DM);
  float var  = ss * (1.f / DM) - mean * mean;
  float rs   = rsqrtf(var + 1e-5f);
  for (int d = 0; d < DM; ++d) {
    float v = xb[(size_t)d * LTOK] + lb[d];
    out[((size_t)b * DM + d) * LTOK + l] = (v - mean) * rs * w[d] + bias[d];
  }
}

// -------------------- f32 WMMA GEMM:  C = A * W^T --------------------------
// A: [M,K] lda;  W: [N,K] ldw (B[k][n] = W[n*ldw+k]);  C: [M,N] ldc.
// One wave per 16x16 C tile, K stepped by 4 via V_WMMA_F32_16X16X4_F32.
// Ragged N handled branch-free: clamp the load row, zero by multiply
// (v_cndmask, no EXEC churn around the WMMA).
// epilogue: 0 = none, 1 = softplus(v + bias[n]).
__global__ __launch_bounds__(128) void gemm_wmma_f32(
    const float* __restrict__ A, int lda,
    const float* __restrict__ W, int ldw,
    float* __restrict__ C, int ldc,
    int M, int N, int K, int epilogue, const float* __restrict__ bias) {
  int lane = threadIdx.x & 31;
  int wave = threadIdx.x >> 5;
  int tn = blockIdx.x;
  int tm = blockIdx.y * 4 + wave;
  int row0 = tm * 16, col0 = tn * 16;
  if (row0 >= M) return;                 // wave-uniform
  int half = lane >> 4;                  // 0: lanes 0-15, 1: lanes 16-31
  int idx  = lane & 15;
  int mrow = row0 + idx;                 // A row held by this lane
  int ncol = col0 + idx;                 // B col held by this lane
  bool nvalid = (ncol < N);
  int ncl = nvalid ? ncol : (N - 1);     // clamped: always a legal address
  float msk = nvalid ? 1.f : 0.f;
  // frag layout: VGPR v, half h -> K = k0 + 2h + v
  const float* Arow = A + (size_t)mrow * lda + 2 * half;
  const float* Wrow = W + (size_t)ncl  * ldw + 2 * half;
  v8f acc = {};
#pragma unroll 2
  for (int k0 = 0; k0 < K; k0 += 4) {
    v2f a  = *reinterpret_cast<const v2f*>(Arow + k0);
    v2f bf = *reinterpret_cast<const v2f*>(Wrow + k0);
    bf.x *= msk; bf.y *= msk;
    acc = __builtin_amdgcn_wmma_f32_16x16x4_f32(
        false, a, false, bf, (short)0, acc, false, false);
  }
  // C/D layout: VGPR v -> M = row0 + 8*half + v, N = col0 + idx
#pragma unroll
  for (int v = 0; v < 8; ++v) {
    int r = row0 + half * 8 + v;
    int c = col0 + idx;
    if (c < N) {
      float val = acc[v];
      if (epilogue == 1) {
        val += bias[c];
        val = (val > 20.f) ? val : log1pf(__expf(val));
      }
      C[(size_t)r * ldc + c] = val;
    }
  }
}

// --------------- depthwise causal conv (k=4) + SiLU ------------------------
__global__ __launch_bounds__(256) void conv_silu_kernel(
    const float* __restrict__ xz, const float* __restrict__ cw,
    const float* __restrict__ cb, float* __restrict__ xc, int rev) {
  int m = blockIdx.x;          // direction-local token index (0..BL-1)
  int d = threadIdx.x;         // channel
  int b = m >> 13, l = m & (LTOK - 1);
  float acc = cb[d];
#pragma unroll
  for (int j = 0; j < 4; ++j) {
    int ls = l - 3 + j;
    if (ls >= 0) {
      int lg = rev ? (LTOK - 1 - ls) : ls;
      acc += cw[d * 4 + j] * xz[((size_t)(b << 13) + lg) * 512 + d];
    }
  }
  xc[(size_t)m * DI + d] = acc / (1.f + __expf(-acc));   // silu
}

// ----------------- chunked selective scan: pass 1 (local) ------------------
// gw = (b, d-pair, chunk). Lanes 0-15 -> states of d0, 16-31 -> d1.
// Produces h0(c) (chunk scan from h_in=0) and sum(dt) over the chunk.
__global__ __launch_bounds__(128) void scan_part1_kernel(
    const float* __restrict__ dt, const float* __restrict__ dbl,
    const float* __restrict__ xc, const float* __restrict__ A_log,
    float* __restrict__ h0buf, float* __restrict__ sbuf) {
  int lane = threadIdx.x & 31;
  int wv   = threadIdx.x >> 5;
  int gw   = blockIdx.x * 4 + wv;         // 0..8191
  int c    = gw & (NCH - 1);
  int rest = gw >> 4;                     // 0..511
  int b    = rest >> 7;
  int half = lane >> 4;
  int n    = lane & 15;
  int d    = ((rest & 127) << 1) + half;
  float An = -__expf(A_log[d * NSTATE + n]);
  size_t tok0 = ((size_t)b << 13) + (size_t)c * CHLEN;
  float h = 0.f, sdt = 0.f;
  for (int t = 0; t < CHLEN; ++t) {
    size_t tok = tok0 + t;
    float dt_t = dt[tok * DI + d];
    float x_t  = xc[tok * DI + d];
    float B_t  = dbl[tok * NPROJ + 8 + n];
    h = __expf(dt_t * An) * h + (dt_t * x_t) * B_t;
    sdt += dt_t;
  }
  size_t hidx = (((size_t)b * DI + d) * NCH + c) * NSTATE + n;
  h0buf[hidx] = h;
  if (n == 0) sbuf[((size_t)b * DI + d) * NCH + c] = sdt;
}

// -------------- chunked scan: pass 2 (serial chunk combine) ----------------
// One thread per (b,d,n): h_in(c) = h0(c-1) + exp(A*Sdt(c-1)) * h_in(c-1).
__global__ __launch_bounds__(256) void scan_part2_kernel(
    const float* __restrict__ h0buf, const float* __restrict__ sbuf,
    const float* __restrict__ A_log, float* __restrict__ hinbuf) {
  int g  = blockIdx.x * blockDim.x + threadIdx.x;  // 0..16383
  int n  = g & (NSTATE - 1);
  int bd = g >> 4;                                 // b*DI + d
  int d  = bd & (DI - 1);
  float An = -__expf(A_log[d * NSTATE + n]);
  float h = 0.f;
  for (int c = 0; c < NCH; ++c) {
    size_t idx = ((size_t)bd * NCH + c) * NSTATE + n;
    hinbuf[idx] = h;
    float P = __expf(An * sbuf[(size_t)bd * NCH + c]);
    h = h0buf[idx] + P * h;
  }
}

// --------- chunked scan: pass 3 (re-scan from exact h_in, emit y) ----------
// y_t = sum_n h*C_n (shfl_xor reduce) ; y = (y_t + xc*D) * silu(z).
// Backward dir (rev=1) reads z / writes y at reversed token; accumulates.
__global__ __launch_bounds__(128) void scan_part3_kernel(
    const float* __restrict__ dt, const float* __restrict__ dbl,
    const float* __restrict__ xc, const float* __restrict__ xz,
    const float* __restrict__ A_log, const float* __restrict__ Dv,
    const float* __restrict__ hinbuf, float* __restrict__ y,
    int rev, int accum) {
  int lane = threadIdx.x & 31;
  int wv   = threadIdx.x >> 5;
  int gw   = blockIdx.x * 4 + wv;         // 0..8191
  int c    = gw & (NCH - 1);
  int rest = gw >> 4;
  int b    = rest >> 7;
  int half = lane >> 4;
  int n    = lane & 15;
  int d    = ((rest & 127) << 1) + half;
  float An    = -__expf(A_log[d * NSTATE + n]);
  float Dskip = Dv[d];
  float h = hinbuf[(((size_t)b * DI + d) * NCH + c) * NSTATE + n];
  size_t tokbase = (size_t)b << 13;
  size_t tok0 = tokbase + (size_t)c * CHLEN;
  for (int t = 0; t < CHLEN; ++t) {
    size_t tok = tok0 + t;
    float dt_t = dt[tok * DI + d];
    float x_t  = xc[tok * DI + d];
    float B_t  = dbl[tok * NPROJ + 8 + n];
    float C_t  = dbl[tok * NPROJ + 24 + n];
    h = __expf(dt_t * An) * h + (dt_t * x_t) * B_t;
    float p = h * C_t;
    p += __shfl_xor(p, 1);
    p += __shfl_xor(p, 2);
    p += __shfl_xor(p, 4);
    p += __shfl_xor(p, 8);
    if (n == 0) {
      int tl = c * CHLEN + t;                    // direction-local token
      int tg = rev ? (LTOK - 1 - tl) : tl;       // global token
      size_t tokg = tokbase + tg;
      float zv = xz[tokg * 512 + DI + d];
      float g  = zv / (1.f + __expf(-zv));
      float val = (p + x_t * Dskip) * g;
      float* yp = y + tokg * DI + d;
      if (accum) *yp += val; else *yp = val;
    }
  }
}

// ---------------------------------------------------------------------------
extern "C" void kernel_launch(void* const* d_in, const int* in_sizes, int n_in,
                              void* d_out, int out_size, void* d_ws, size_t ws_size,
                              hipStream_t stream) {
  const float* x         = (const float*)d_in[0];
  const float* ln1_w     = (const float*)d_in[1];
  const float* ln1_b     = (const float*)d_in[2];
  const float* ln2_w     = (const float*)d_in[3];
  const float* ln2_b     = (const float*)d_in[4];
  const float* in_proj_w = (const float*)d_in[5];
  const float* out_proj_w= (const float*)d_in[6];
  const float* conv_w_f  = (const float*)d_in[7];
  const float* conv_b_f  = (const float*)d_in[8];
  const float* xproj_w_f = (const float*)d_in[9];
  const float* dtproj_w_f= (const float*)d_in[10];
  const float* dtproj_b_f= (const float*)d_in[11];
  const float* A_log_f   = (const float*)d_in[12];
  const float* D_f       = (const float*)d_in[13];
  const float* conv_w_b  = (const float*)d_in[14];
  const float* conv_b_b  = (const float*)d_in[15];
  const float* xproj_w_b = (const float*)d_in[16];
  const float* dtproj_w_b= (const float*)d_in[17];
  const float* dtproj_b_b= (const float*)d_in[18];
  const float* A_log_b   = (const float*)d_in[19];
  const float* D_b       = (const float*)d_in[20];
  float* out = (float*)d_out;

  // workspace layout (floats)
  float* ws = (float*)d_ws;
  float* xz     = ws;                          // BL*512
  float* xn     = xz  + (size_t)BL * 512;      // BL*128
  float* xc     = xn  + (size_t)BL * 128;      // BL*256 (reused per dir)
  float* dbl    = xc  + (size_t)BL * 256;      // BL*40  (reused per dir)
  float* dtb    = dbl + (size_t)BL * 40;       // BL*256 (reused per dir)
  float* ybuf   = dtb + (size_t)BL * 256;      // BL*256 (y_f + y_b)
  float* outlin = ybuf+ (size_t)BL * 256;      // BL*128
  float* h0buf  = outlin + (size_t)BL * 128;   // B*DI*NCH*NSTATE = 262144
  float* hinbuf = h0buf + (size_t)BATCH*DI*NCH*NSTATE;
  float* sbuf   = hinbuf+ (size_t)BATCH*DI*NCH*NSTATE;  // B*DI*NCH = 16384

  // 1) LN1
  ln_in_kernel<<<BL / 256, 256, 0, stream>>>(x, ln1_w, ln1_b, xn);

  // 2) in_proj: [BL,128] @ [128,512]^T -> xz [BL,512]
  gemm_wmma_f32<<<dim3(512 / 16, BL / 64), 128, 0, stream>>>(
      xn, DM, in_proj_w, DM, xz, 512, BL, 512, DM, 0, nullptr);

  // 3) forward + backward directions (sequential; share xc/dbl/dt buffers)
  for (int dir = 0; dir < 2; ++dir) {
    int rev = dir;
    const float* cw  = dir ? conv_w_b   : conv_w_f;
    const float* cb  = dir ? conv_b_b   : conv_b_f;
    const float* xpw = dir ? xproj_w_b  : xproj_w_f;
    const float* dpw = dir ? dtproj_w_b : dtproj_w_f;
    const float* dpb = dir ? dtproj_b_b : dtproj_b_f;
    const float* Al  = dir ? A_log_b    : A_log_f;
    const float* Dv  = dir ? D_b        : D_f;

    conv_silu_kernel<<<BL, 256, 0, stream>>>(xz, cw, cb, xc, rev);

    // x_proj: [BL,256] @ [256,40]^T -> dbl [BL,40]  (3 N-tiles, guarded)
    gemm_wmma_f32<<<dim3(3, BL / 64), 128, 0, stream>>>(
        xc, DI, xpw, DI, dbl, NPROJ, BL, NPROJ, DI, 0, nullptr);

    // dt_proj + softplus: [BL,8] @ [8,256]^T -> dt [BL,256]
    gemm_wmma_f32<<<dim3(256 / 16, BL / 64), 128, 0, stream>>>(
        dbl, NPROJ, dpw, 8, dtb, DI, BL, DI, 8, 1, dpb);

    // chunked selective scan
    scan_part1_kernel<<<2048, 128, 0, stream>>>(dtb, dbl, xc, Al, h0buf, sbuf);
    scan_part2_kernel<<<64, 256, 0, stream>>>(h0buf, sbuf, Al, hinbuf);
    scan_part3_kernel<<<2048, 128, 0, stream>>>(
        dtb, dbl, xc, xz, Al, Dv, hinbuf, ybuf, rev, dir);
  }

  // 4) out_proj: (y_f+y_b)[BL,256] @ [256,128]^T -> outlin [BL,128]
  gemm_wmma_f32<<<dim3(DM / 16, BL / 64), 128, 0, stream>>>(
      ybuf, DI, out_proj_w, DI, outlin, DM, BL, DM, DI, 0, nullptr);

  // 5) residual + LN2 -> d_out [B,128,L]
  ln_out_kernel<<<BL / 256, 256, 0, stream>>>(x, outlin, ln2_w, ln2_b, out);
}